// SpatialTransformerPure_45578192945735
// MI455X (gfx1250) — compile-verified
//
#include <hip/hip_runtime.h>

// Problem dims (fixed by the reference's setup_inputs).
#define B_ 2
#define C_ 2
#define D_ 160
#define H_ 192
#define W_ 224

// One output row per block: blockDim.x = W = 224 = 7 wave32 waves.
// grid = (H, D, B). w = tid, h/d/b come from SGPR blockIdx -> all row/plane
// base offsets are scalar (SALU) and shared per wave; no integer division.
//
// Cache policy (MI455X: 192 MB L2; vol = 110 MB fits resident):
//   - phi: nontemporal loads  (streamed once, 165 MB, don't pollute L2)
//   - out: nontemporal stores (streamed once, 110 MB)
//   - vol: regular RT gathers (want the whole volume L2-resident)
//   - global_prefetch_b8 at the identity position (displacements ~N(0,3)
//     voxels, so the identity cacheline is nearly always one of the 8 corner
//     lines), issued before the phi load -> address -> gather chain.
__global__ __launch_bounds__(W_) void warp3d_trilinear_kernel(
    const float* __restrict__ vol,   // (B, C, D, H, W)
    const float* __restrict__ phi,   // (B, 3, D, H, W), displacement in voxels
    float* __restrict__ out)         // (B, C, D, H, W)
{
    const int HW  = H_ * W_;          // 43,008
    const int DHW = D_ * HW;          // 6,881,280  (fits int; *4B = 27.5 MB)

    const int w = threadIdx.x;
    const int h = blockIdx.x;
    const int d = blockIdx.y;
    const int b = blockIdx.z;

    const int r = d * HW + h * W_ + w;            // intra-batch spatial offset

    const float* vb = vol + (long)b * (C_ * DHW); // batch base of vol
    float*       ob = out + (long)b * (C_ * DHW) + r;

    // Prefetch identity-position voxel lines for both channels before the
    // dependent phi load starts.
    __builtin_prefetch(vb + r,        0, 3);
    __builtin_prefetch(vb + DHW + r,  0, 3);

    // phi components: [0]->depth, [1]->height, [2]->width displacement.
    const float* pb = phi + (long)b * (3 * DHW) + r;
    const float px = __builtin_nontemporal_load(pb);
    const float py = __builtin_nontemporal_load(pb + DHW);
    const float pz = __builtin_nontemporal_load(pb + 2 * DHW);

    // Border padding: clamp sample coordinate into the volume.
    const float cd = fminf(fmaxf((float)d + px, 0.0f), (float)(D_ - 1));
    const float ch = fminf(fmaxf((float)h + py, 0.0f), (float)(H_ - 1));
    const float cw = fminf(fmaxf((float)w + pz, 0.0f), (float)(W_ - 1));

    // cd/ch/cw >= 0 after clamp, so int truncation == floor.
    const int d0 = (int)cd;  const int d1 = min(d0 + 1, D_ - 1);
    const int h0 = (int)ch;  const int h1 = min(h0 + 1, H_ - 1);
    const int w0 = (int)cw;  const int w1 = min(w0 + 1, W_ - 1);

    const float fd = cd - (float)d0;
    const float fh = ch - (float)h0;
    const float fw = cw - (float)w0;
    const float gd = 1.0f - fd;
    const float gh = 1.0f - fh;
    const float gw = 1.0f - fw;

    const float k000 = gd * gh * gw;
    const float k001 = gd * gh * fw;
    const float k010 = gd * fh * gw;
    const float k011 = gd * fh * fw;
    const float k100 = fd * gh * gw;
    const float k101 = fd * gh * fw;
    const float k110 = fd * fh * gw;
    const float k111 = fd * fh * fw;

    // Four (d,h) plane-row offsets shared across the 8 corners (32-bit math).
    const int o00 = d0 * HW + h0 * W_;
    const int o01 = d0 * HW + h1 * W_;
    const int o10 = d1 * HW + h0 * W_;
    const int o11 = d1 * HW + h1 * W_;

#pragma unroll
    for (int c = 0; c < C_; ++c) {
        const float* vc = vb + c * DHW;
        float acc;
        acc  = vc[o00 + w0] * k000;
        acc  = fmaf(vc[o00 + w1], k001, acc);
        acc  = fmaf(vc[o01 + w0], k010, acc);
        acc  = fmaf(vc[o01 + w1], k011, acc);
        acc  = fmaf(vc[o10 + w0], k100, acc);
        acc  = fmaf(vc[o10 + w1], k101, acc);
        acc  = fmaf(vc[o11 + w0], k110, acc);
        acc  = fmaf(vc[o11 + w1], k111, acc);
        __builtin_nontemporal_store(acc, ob + c * DHW);
    }
}

extern "C" void kernel_launch(void* const* d_in, const int* in_sizes, int n_in,
                              void* d_out, int out_size, void* d_ws, size_t ws_size,
                              hipStream_t stream) {
    (void)in_sizes; (void)n_in; (void)out_size; (void)d_ws; (void)ws_size;
    const float* vol = (const float*)d_in[0];  // (B,C,D,H,W) float32
    const float* phi = (const float*)d_in[1];  // (B,3,D,H,W) float32
    float* out = (float*)d_out;                // (B,C,D,H,W) float32

    dim3 block(W_, 1, 1);          // 224 threads = 7 wave32 waves, one row
    dim3 grid(H_, D_, B_);         // 192 x 160 x 2 blocks

    warp3d_trilinear_kernel<<<grid, block, 0, stream>>>(vol, phi, out);
}